// SSM_90726889160767
// MI455X (gfx1250) — compile-verified
//
#include <hip/hip_runtime.h>
#include <stdint.h>

#define L_SEQ  4096
#define DMODEL 2048
#define NSTATE 16
#define RLOW   128
#define NPROJ  160   // r + 2n
#define NCHUNK 64
#define TCHUNK 64    // L_SEQ / NCHUNK

typedef __attribute__((ext_vector_type(2))) float v2f;
typedef __attribute__((ext_vector_type(4))) float v4f;
typedef __attribute__((ext_vector_type(8))) float v8f;

// Async copy 16B global -> LDS (CDNA5 GLOBAL_LOAD_ASYNC_TO_LDS_B128, ASYNCcnt).
// lds generic pointer's low 32 bits are the LDS byte address (aperture layout).
__device__ __forceinline__ void async_g2l_b128(const float* gsrc, float* lds_dst) {
    uint32_t ldsaddr = (uint32_t)(uintptr_t)lds_dst;
    uint64_t gaddr   = (uint64_t)(uintptr_t)gsrc;
    asm volatile("global_load_async_to_lds_b128 %0, %1, off"
                 :: "v"(ldsaddr), "v"(gaddr) : "memory");
}
__device__ __forceinline__ void wait_async0() {
    asm volatile("s_wait_asynccnt 0x0" ::: "memory");
}

// ---------------------------------------------------------------------------
// Kernel 1: proj = x @ W_xproj^T   (L x 160), fp32 WMMA 16x16x4.
// One wave per 16-row M tile; computes all 10 N tiles so x is read once and
// the A fragment is reused 10x. W_xproj (1.3MB) stays resident in L2.
// ---------------------------------------------------------------------------
__global__ __launch_bounds__(32)
void proj_gemm(const float* __restrict__ x, const float* __restrict__ W,
               float* __restrict__ proj) {
    const int m0   = blockIdx.x * 16;
    const int tl   = threadIdx.x;          // 0..31
    const int l16  = tl & 15;              // A row (M) / B col (N) within tile
    const int kp   = (tl >> 4) * 2;        // K sub-pair selected by lane half

    v8f acc[10];
    #pragma unroll
    for (int t = 0; t < 10; ++t) acc[t] = (v8f)(0.0f);

    const float* xrow = x + (size_t)(m0 + l16) * DMODEL;
    for (int k = 0; k < DMODEL; k += 4) {
        v2f a;
        a.x = xrow[k + kp];
        a.y = xrow[k + kp + 1];
        #pragma unroll
        for (int t = 0; t < 10; ++t) {
            const int nn = t * 16 + l16;   // output column (row of W_xproj)
            v2f b;
            b.x = W[(size_t)nn * DMODEL + k + kp];
            b.y = W[(size_t)nn * DMODEL + k + kp + 1];
            acc[t] = __builtin_amdgcn_wmma_f32_16x16x4_f32(
                false, a, false, b, (short)0, acc[t], false, false);
        }
    }

    // D layout: VGPR v holds row (v + 8*(lane>=16)), col = lane&15
    const int rbase = (tl >> 4) * 8;
    #pragma unroll
    for (int t = 0; t < 10; ++t)
        #pragma unroll
        for (int v = 0; v < 8; ++v)
            proj[(size_t)(m0 + v + rbase) * NPROJ + t * 16 + l16] = acc[t][v];
}

// ---------------------------------------------------------------------------
// Kernel 2: dt = dt_low @ W_dt^T + b_dt   (L x 2048), fp32 WMMA, K=128.
// dt_low lives inside proj (columns 16..143). One wave per 16x16 tile.
// dt is re-read by both scan passes -> keep RT stores (L2 resident, 33MB).
// ---------------------------------------------------------------------------
__global__ __launch_bounds__(32)
void dt_gemm(const float* __restrict__ proj, const float* __restrict__ Wdt,
             const float* __restrict__ bdt, float* __restrict__ dt) {
    const int m0  = blockIdx.x * 16;       // L tile
    const int n0  = blockIdx.y * 16;       // d tile
    const int tl  = threadIdx.x;
    const int l16 = tl & 15;
    const int kp  = (tl >> 4) * 2;

    v8f acc = (v8f)(0.0f);
    const float* arow = proj + (size_t)(m0 + l16) * NPROJ + NSTATE; // dt_low row
    const float* brow = Wdt  + (size_t)(n0 + l16) * RLOW;           // W_dt row (B col)
    #pragma unroll
    for (int k = 0; k < RLOW; k += 4) {
        v2f a, b;
        a.x = arow[k + kp]; a.y = arow[k + kp + 1];
        b.x = brow[k + kp]; b.y = brow[k + kp + 1];
        acc = __builtin_amdgcn_wmma_f32_16x16x4_f32(
            false, a, false, b, (short)0, acc, false, false);
    }

    const float bias = bdt[n0 + l16];
    const int rbase = (tl >> 4) * 8;
    #pragma unroll
    for (int v = 0; v < 8; ++v)
        dt[(size_t)(m0 + v + rbase) * DMODEL + n0 + l16] = acc[v] + bias;
}

// ---------------------------------------------------------------------------
// Kernel 3 (scan pass 1): per chunk c, per channel dd: chunk product
// P = prod(da) and chunk-local scan end S (h0 = 0).
// B rows staged in LDS via async b128 copies (1 op/lane = 4KB/block).
// ---------------------------------------------------------------------------
__global__ __launch_bounds__(256)
void scan_pass1(const float* __restrict__ dt, const float* __restrict__ x,
                const float* __restrict__ proj, const float* __restrict__ A,
                float* __restrict__ P, float* __restrict__ S) {
    __shared__ float Bsh[TCHUNK * NSTATE];
    const int c  = blockIdx.y;
    const int dd = blockIdx.x * 256 + threadIdx.x;
    const int l0 = c * TCHUNK;

    {   // async stage: B = proj[l0:l0+64, 0:16]  (row stride 640B, 16B aligned)
        const int lr = threadIdx.x >> 2;           // 0..63
        const int j4 = (threadIdx.x & 3) * 4;      // 0,4,8,12
        async_g2l_b128(proj + (size_t)(l0 + lr) * NPROJ + j4,
                       &Bsh[lr * NSTATE + j4]);
        wait_async0();
    }
    __syncthreads();

    float areg[NSTATE], p[NSTATE], s[NSTATE];
    #pragma unroll
    for (int j = 0; j < NSTATE; ++j) {
        areg[j] = A[(size_t)dd * NSTATE + j];
        p[j] = 1.0f; s[j] = 0.0f;
    }

    for (int t = 0; t < TCHUNK; ++t) {
        const size_t idx = (size_t)(l0 + t) * DMODEL + dd;
        const float dtv = dt[idx];
        const float dtx = dtv * x[idx];
        #pragma unroll
        for (int j = 0; j < NSTATE; ++j) {
            const float da = dtv * areg[j];
            s[j] = fmaf(s[j], da, dtx * Bsh[t * NSTATE + j]);
            p[j] *= da;
        }
    }

    const size_t base = ((size_t)c * DMODEL + dd) * NSTATE;
    #pragma unroll
    for (int j = 0; j < NSTATE; ++j) { P[base + j] = p[j]; S[base + j] = s[j]; }
}

// ---------------------------------------------------------------------------
// Kernel 4: sequential carry across the 64 chunks (parallel over d*n = 32768).
// H[c] = h entering chunk c;  carry = carry*P[c] + S[c].
// ---------------------------------------------------------------------------
__global__ __launch_bounds__(256)
void chunk_carry(const float* __restrict__ P, const float* __restrict__ S,
                 float* __restrict__ H) {
    const int e = blockIdx.x * 256 + threadIdx.x;  // 0 .. d*n-1
    float carry = 0.0f;
    for (int c = 0; c < NCHUNK; ++c) {
        const size_t idx = (size_t)c * (DMODEL * NSTATE) + e;
        H[idx] = carry;
        carry = fmaf(carry, P[idx], S[idx]);
    }
}

// ---------------------------------------------------------------------------
// Kernel 5 (scan pass 2): redo each chunk's scan with the correct incoming h,
// write hs (L,d,n) and y = sum_n hs*C + x*D.
// hs/y are write-once-never-read (570MB total): non-temporal stores so they
// don't evict the reused dt/proj working set from L2.
// ---------------------------------------------------------------------------
__global__ __launch_bounds__(256)
void scan_pass2(const float* __restrict__ dt, const float* __restrict__ x,
                const float* __restrict__ proj, const float* __restrict__ A,
                const float* __restrict__ D, const float* __restrict__ H,
                float* __restrict__ y, float* __restrict__ hs) {
    __shared__ float Bsh[TCHUNK * NSTATE];
    __shared__ float Csh[TCHUNK * NSTATE];
    const int c  = blockIdx.y;
    const int dd = blockIdx.x * 256 + threadIdx.x;
    const int l0 = c * TCHUNK;

    {   // async stage: B = proj[:, 0:16], C = proj[:, 144:160]
        const int lr = threadIdx.x >> 2;
        const int j4 = (threadIdx.x & 3) * 4;
        const float* prow = proj + (size_t)(l0 + lr) * NPROJ;
        async_g2l_b128(prow + j4,                  &Bsh[lr * NSTATE + j4]);
        async_g2l_b128(prow + NSTATE + RLOW + j4,  &Csh[lr * NSTATE + j4]);
        wait_async0();
    }
    __syncthreads();

    float areg[NSTATE], h[NSTATE];
    const size_t hbase = ((size_t)c * DMODEL + dd) * NSTATE;
    #pragma unroll
    for (int j = 0; j < NSTATE; ++j) {
        areg[j] = A[(size_t)dd * NSTATE + j];
        h[j]    = H[hbase + j];
    }
    const float dcoef = D[dd];

    for (int t = 0; t < TCHUNK; ++t) {
        const size_t idx = (size_t)(l0 + t) * DMODEL + dd;
        const float dtv = dt[idx];
        const float xv  = x[idx];
        const float dtx = dtv * xv;
        float ysum = 0.0f;
        #pragma unroll
        for (int j = 0; j < NSTATE; ++j) {
            const float da = dtv * areg[j];
            h[j] = fmaf(h[j], da, dtx * Bsh[t * NSTATE + j]);
            ysum = fmaf(h[j], Csh[t * NSTATE + j], ysum);
        }
        __builtin_nontemporal_store(fmaf(xv, dcoef, ysum), &y[idx]);
        v4f* outp = (v4f*)(hs + idx * NSTATE);   // 64B per lane, coalesced
        #pragma unroll
        for (int q = 0; q < 4; ++q) {
            v4f hv = { h[4*q], h[4*q+1], h[4*q+2], h[4*q+3] };
            __builtin_nontemporal_store(hv, &outp[q]);
        }
    }
}

// ---------------------------------------------------------------------------
extern "C" void kernel_launch(void* const* d_in, const int* in_sizes, int n_in,
                              void* d_out, int out_size, void* d_ws, size_t ws_size,
                              hipStream_t stream) {
    const float* x       = (const float*)d_in[0];
    const float* W_xproj = (const float*)d_in[1];
    const float* W_dt    = (const float*)d_in[2];
    const float* b_dt    = (const float*)d_in[3];
    const float* A       = (const float*)d_in[4];
    const float* D       = (const float*)d_in[5];

    float* ws   = (float*)d_ws;
    float* proj = ws;                                          // L*160
    float* dt   = proj + (size_t)L_SEQ * NPROJ;                // L*d
    float* P    = dt   + (size_t)L_SEQ * DMODEL;               // C*d*n
    float* S    = P    + (size_t)NCHUNK * DMODEL * NSTATE;     // C*d*n
    float* H    = S    + (size_t)NCHUNK * DMODEL * NSTATE;     // C*d*n

    float* y  = (float*)d_out;
    float* hs = y + (size_t)L_SEQ * DMODEL;   // outputs concatenated: (y, hs)

    proj_gemm <<<L_SEQ / 16, 32, 0, stream>>>(x, W_xproj, proj);
    dt_gemm   <<<dim3(L_SEQ / 16, DMODEL / 16), 32, 0, stream>>>(proj, W_dt, b_dt, dt);
    scan_pass1<<<dim3(DMODEL / 256, NCHUNK), 256, 0, stream>>>(dt, x, proj, A, P, S);
    chunk_carry<<<(DMODEL * NSTATE) / 256, 256, 0, stream>>>(P, S, H);
    scan_pass2<<<dim3(DMODEL / 256, NCHUNK), 256, 0, stream>>>(dt, x, proj, A, D, H, y, hs);
}